// MultichannelBeamSearch_61495341744238
// MI455X (gfx1250) — compile-verified
//
#include <hip/hip_runtime.h>

#define BSZ   32
#define BEAM  4
#define VOCAB 50257
#define K     8
#define NT    1024   // 32 waves per block

typedef __attribute__((ext_vector_type(2))) float v2f;
typedef __attribute__((ext_vector_type(8))) float v8f;

// ---------------------------------------------------------------------------
// Sorted-descending top-8 register list: insert with early-out.
// Static indices only (fully unrolled) -> stays in VGPRs.
// ---------------------------------------------------------------------------
__device__ __forceinline__ void tk_insert(float (&v)[K], int (&ix)[K], float x, int i) {
    if (x <= v[K - 1]) return;
    v[K - 1] = x; ix[K - 1] = i;
#pragma unroll
    for (int p = K - 1; p > 0; --p) {
        if (v[p] > v[p - 1]) {
            float tv = v[p]; v[p] = v[p - 1]; v[p - 1] = tv;
            int   ti = ix[p]; ix[p] = ix[p - 1]; ix[p - 1] = ti;
        } else {
            break;
        }
    }
}

__device__ __forceinline__ void cswap_desc(float &a, float &b, int &ai, int &bi) {
    if (b > a) {
        float t = a; a = b; b = t;
        int  ti = ai; ai = bi; bi = ti;
    }
}

// Merge two sorted-descending 8-lists -> top-8 sorted descending, into (a, ai).
// Bitonic trick: top-8 multiset = { max(a[k], b[7-k]) }, which is bitonic;
// then 3 static bitonic stages (dist 4,2,1). No dynamic register indexing.
__device__ __forceinline__ void merge8(float (&a)[K], int (&ai)[K],
                                       const float (&b)[K], const int (&bi)[K]) {
    float m[K]; int mi[K];
#pragma unroll
    for (int k = 0; k < K; ++k) {
        bool ta = a[k] >= b[K - 1 - k];
        m[k]  = ta ? a[k]  : b[K - 1 - k];
        mi[k] = ta ? ai[k] : bi[K - 1 - k];
    }
#pragma unroll
    for (int k = 0; k < 4; ++k) cswap_desc(m[k], m[k + 4], mi[k], mi[k + 4]);
    cswap_desc(m[0], m[2], mi[0], mi[2]);
    cswap_desc(m[1], m[3], mi[1], mi[3]);
    cswap_desc(m[4], m[6], mi[4], mi[6]);
    cswap_desc(m[5], m[7], mi[5], mi[7]);
#pragma unroll
    for (int k = 0; k < K; k += 2) cswap_desc(m[k], m[k + 1], mi[k], mi[k + 1]);
#pragma unroll
    for (int k = 0; k < K; ++k) { a[k] = m[k]; ai[k] = mi[k]; }
}

// 5-level shuffle-xor merge across a wave32: all lanes converge to the wave top-8.
__device__ __forceinline__ void wave_merge8(float (&v)[K], int (&ix)[K]) {
#pragma unroll
    for (int off = 1; off < 32; off <<= 1) {
        float bv[K]; int bi[K];
#pragma unroll
        for (int k = 0; k < K; ++k) {
            bv[k] = __shfl_xor(v[k],  off, 32);
            bi[k] = __shfl_xor(ix[k], off, 32);
        }
        merge8(v, ix, bv, bi);
    }
}

// Fast-path top-8 update for 4 elements: single max-tree pre-check.
__device__ __forceinline__ void tk_insert4(float (&v)[K], int (&ix)[K],
                                           float4 x, int base) {
    float m = fmaxf(fmaxf(x.x, x.y), fmaxf(x.z, x.w));
    if (m > v[K - 1]) {
        tk_insert(v, ix, x.x, base);
        tk_insert(v, ix, x.y, base + 1);
        tk_insert(v, ix, x.z, base + 2);
        tk_insert(v, ix, x.w, base + 3);
    }
}

// ---------------------------------------------------------------------------
// Kernel A: per-(b,beam,channel) row top-8 over VOCAB. 256 blocks x 1024 thr
// (32 waves). Bandwidth-bound: 4-deep pipelined b128 coalesced loads +
// prefetch; bias added only at the end (constant shift preserves per-row
// top-k order). Reduction: shuffle merges in-wave, 32 leader lists via LDS,
// shuffle merge in wave 0.   blockIdx.x = b*8 + beam*2 + ch
// ---------------------------------------------------------------------------
__global__ __launch_bounds__(NT) void row_topk_kernel(
        const float* __restrict__ lp1, const float* __restrict__ lp2,
        const float* __restrict__ sc1, const float* __restrict__ sc2,
        const int*   __restrict__ stepp,
        float* __restrict__ wv, int* __restrict__ wi) {
    const int bx   = blockIdx.x;
    const int b    = bx >> 3;
    const int beam = (bx >> 1) & 3;
    const int ch   = bx & 1;
    const float* lp = ch ? lp2 : lp1;
    const float* sc = ch ? sc2 : sc1;

    const size_t roff = (size_t)(b * BEAM + beam) * VOCAB;
    const float* row  = lp + roff;
    const int tid = threadIdx.x;

    float v[K]; int ix[K];
#pragma unroll
    for (int k = 0; k < K; ++k) { v[k] = -__builtin_inff(); ix[k] = 0; }

    // Rows are only 4B aligned (50257 % 4 == 1): scalar head to 16B alignment.
    const int head = (int)((4u - ((unsigned)roff & 3u)) & 3u);
    const int n4   = (VOCAB - head) >> 2;
    const float4* row4 = (const float4*)(row + head);

    // 4-deep software-pipelined main loop: 4 independent b128 loads in flight.
    int c = tid;
    for (; c + 3 * NT < n4; c += 4 * NT) {
        if (c + 4 * NT < n4) __builtin_prefetch(&row4[c + 4 * NT], 0, 0);
        float4 x0 = row4[c];
        float4 x1 = row4[c + NT];
        float4 x2 = row4[c + 2 * NT];
        float4 x3 = row4[c + 3 * NT];
        tk_insert4(v, ix, x0, head + (c << 2));
        tk_insert4(v, ix, x1, head + ((c + NT) << 2));
        tk_insert4(v, ix, x2, head + ((c + 2 * NT) << 2));
        tk_insert4(v, ix, x3, head + ((c + 3 * NT) << 2));
    }
    for (; c < n4; c += NT) {
        float4 x = row4[c];
        tk_insert4(v, ix, x, head + (c << 2));
    }
    if (tid == 0) {  // misalignment head + scalar tail (<= 3 + 3 elements)
        for (int i = 0; i < head; ++i)                 tk_insert(v, ix, row[i], i);
        for (int i = head + (n4 << 2); i < VOCAB; ++i) tk_insert(v, ix, row[i], i);
    }

    // Stage 1: register-only merge within each wave.
    wave_merge8(v, ix);

    // Stage 2: 32 wave-leader lists -> LDS -> wave 0 shuffle merge.
    __shared__ float sv[32 * K];
    __shared__ int   si[32 * K];
    const int wid  = tid >> 5;
    const int lane = tid & 31;
    if (lane == 0) {
#pragma unroll
        for (int k = 0; k < K; ++k) { sv[wid * K + k] = v[k]; si[wid * K + k] = ix[k]; }
    }
    __syncthreads();
    if (wid == 0) {
#pragma unroll
        for (int k = 0; k < K; ++k) { v[k] = sv[lane * K + k]; ix[k] = si[lane * K + k]; }
        wave_merge8(v, ix);
        if (lane == 0) {
            const int step  = *stepp;
            const float bias = sc[(b * BEAM + beam) * 5 + (step - 1)];
#pragma unroll
            for (int k = 0; k < K; ++k) {
                wv[bx * K + k] = v[k] + bias;   // == topk(lprobs + bias) values
                wi[bx * K + k] = ix[k];
            }
        }
    }
}

// ---------------------------------------------------------------------------
// Kernel B: per-batch combine. 32 blocks x 32 threads (one full wave).
// s[beam,t1,t2] = v1[beam][t1] + v2[beam][t2] computed with
// V_WMMA_F32_16X16X4_F32: one 16x16 tile covers two beams using K=3 of 4:
//   A[M,0]=v1[2p+(M>>3)][M&7], A[M,1]=[M<8], A[M,2]=[M>=8], A[M,3]=0
//   B[0,n]=1, B[1,n]=v2[2p][n] (n<8), B[2,n]=v2[2p+1][n] (n<8), B[3,n]=0
// -> D[M,n] = v1 + v2 exactly (fp32 RNE, zeros inert). 2 WMMAs = 4 beams.
// Then lane-local sorted-8 + 5-level shfl_xor bitonic merge -> wave top-8.
// ---------------------------------------------------------------------------
__global__ __launch_bounds__(32) void combine_wmma_kernel(
        const float* __restrict__ wv, const int* __restrict__ wi,
        float* __restrict__ out) {
    const int b    = blockIdx.x;
    const int lane = threadIdx.x;

    __shared__ float s_v1[BEAM][K], s_v2[BEAM][K];
    __shared__ int   s_i1[BEAM][K], s_i2[BEAM][K];
    {
        int beam = lane >> 3, k = lane & 7;
        int r1 = (b * 8 + beam * 2 + 0) * K;
        int r2 = (b * 8 + beam * 2 + 1) * K;
        s_v1[beam][k] = wv[r1 + k];  s_i1[beam][k] = wi[r1 + k];
        s_v2[beam][k] = wv[r2 + k];  s_i2[beam][k] = wi[r2 + k];
    }
    __syncthreads();

    const int  M    = lane & 15;          // A-row / D-row (low half)
    const int  n    = lane & 15;          // B/D column for this lane
    const bool hi   = lane >= 16;
    const bool ncol = (lane & 8) == 0;    // n < 8 -> valid column

    v8f d0, d1;
    {
        v2f a, bb;
        v8f c = {};
        // p = 0 : beams 0,1
        a.x  = hi ? ((M >= 8) ? 1.0f : 0.0f) : s_v1[(M >> 3)][M & 7];
        a.y  = hi ? 0.0f : ((M < 8) ? 1.0f : 0.0f);
        bb.x = hi ? (ncol ? s_v2[1][n & 7] : 0.0f) : 1.0f;
        bb.y = hi ? 0.0f : (ncol ? s_v2[0][n & 7] : 0.0f);
        d0 = __builtin_amdgcn_wmma_f32_16x16x4_f32(false, a, false, bb,
                                                   (short)0, c, false, false);
        // p = 1 : beams 2,3
        a.x  = hi ? ((M >= 8) ? 1.0f : 0.0f) : s_v1[2 + (M >> 3)][M & 7];
        bb.x = hi ? (ncol ? s_v2[3][n & 7] : 0.0f) : 1.0f;
        bb.y = hi ? 0.0f : (ncol ? s_v2[2][n & 7] : 0.0f);
        d1 = __builtin_amdgcn_wmma_f32_16x16x4_f32(false, a, false, bb,
                                                   (short)0, c, false, false);
    }

    // Lane-local sorted top-8 of this lane's 16 tile entries.
    // D VGPR r holds D[M = r + 8*hi, N = n]; beam = 2p+hi, t1 = r, t2 = n.
    float v[K]; int ix[K];
#pragma unroll
    for (int k = 0; k < K; ++k) { v[k] = -__builtin_inff(); ix[k] = 0; }
    const float NEG = -__builtin_inff();
#pragma unroll
    for (int r = 0; r < 8; ++r) {
        float x0 = ncol ? d0[r] : NEG;
        float x1 = ncol ? d1[r] : NEG;
        int  c0  = (((hi ? 1 : 0)    ) << 6) | (r << 3) | (n & 7);
        int  c1  = (((hi ? 1 : 0) + 2) << 6) | (r << 3) | (n & 7);
        tk_insert(v, ix, x0, c0);
        tk_insert(v, ix, x1, c1);
    }

    // Wave-wide merge: all lanes converge to the global top-8.
    wave_merge8(v, ix);

    if (lane < K) {
        int code = ix[lane];
        int beam = code >> 6;
        int t1   = (code >> 3) & 7;
        int t2   = code & 7;
        int o    = b * K + lane;
        out[0 * BSZ * K + o] = s_v1[beam][t1];          // sbuf1
        out[1 * BSZ * K + o] = s_v2[beam][t2];          // sbuf2
        out[2 * BSZ * K + o] = (float)s_i1[beam][t1];   // idx1
        out[3 * BSZ * K + o] = (float)s_i2[beam][t2];   // idx2
        out[4 * BSZ * K + o] = (float)beam;             // beams
    }
}

// ---------------------------------------------------------------------------
extern "C" void kernel_launch(void* const* d_in, const int* in_sizes, int n_in,
                              void* d_out, int out_size, void* d_ws, size_t ws_size,
                              hipStream_t stream) {
    (void)in_sizes; (void)n_in; (void)out_size; (void)ws_size;
    const float* lp1   = (const float*)d_in[0];
    const float* lp2   = (const float*)d_in[1];
    const float* sc1   = (const float*)d_in[2];
    const float* sc2   = (const float*)d_in[3];
    const int*   stepp = (const int*)d_in[4];

    float* wv = (float*)d_ws;                                   // 256*8 floats
    int*   wi = (int*)((char*)d_ws + 256 * K * sizeof(float));  // 256*8 ints

    row_topk_kernel<<<BSZ * BEAM * 2, NT, 0, stream>>>(lp1, lp2, sc1, sc2, stepp, wv, wi);
    combine_wmma_kernel<<<BSZ, 32, 0, stream>>>(wv, wi, (float*)d_out);
}